// TangentSpace_17652315587285
// MI455X (gfx1250) — compile-verified
//
#include <hip/hip_runtime.h>
#include <hip/hip_bf16.h>

#define NCDIM   64
#define LSTR    68          // padded LDS row stride: 68 mod 64 == 4 -> conflict-free strided reads
#define NELEM   (NCDIM*NCDIM)
#define NTRI    2080        // 64*65/2
#define EPSV    1e-6f
#define NS_ITERS    8
#define SQRT_STAGES 4       // 2^4 = 16 scaling factor

typedef float v2f __attribute__((ext_vector_type(2)));
typedef float v8f __attribute__((ext_vector_type(8)));

// ---------------------------------------------------------------------------
// Fragment layouts per CDNA5 ISA 7.12.2 (wave32), V_WMMA_F32_16X16X4_F32:
//   A 16x4 : lane L -> m = L%16, k = 2*(L/16)+v   (v = element 0/1)
//   B 4x16 : lane L -> n = L%16, k = 2*(L/16)+v
//   C 16x16: lane L, vgpr r -> n = L%16, m = r + 8*(L/16)
// All GEMMs: reads complete fully before the mid barrier, so D may alias
// A and/or B (in-place updates are safe).
// ---------------------------------------------------------------------------

// Full-block 64x64x64 GEMM: 8 waves x 2 tiles.  D = alpha*(A@opB(B)) + beta*I
template <bool TRANSB>
__device__ __forceinline__ void gemm64(float* __restrict__ D,
                                       const float* __restrict__ A,
                                       const float* __restrict__ B,
                                       float alpha, float beta_diag)
{
    __syncthreads();
    const int lane = threadIdx.x & 31;
    const int wave = threadIdx.x >> 5;
    const int half = lane >> 4;
    const int l16  = lane & 15;

    const int tm  = wave >> 1;
    const int tn0 = (wave & 1) * 2;
    const int tn1 = tn0 + 1;

    v8f acc0 = {0.f,0.f,0.f,0.f,0.f,0.f,0.f,0.f};
    v8f acc1 = {0.f,0.f,0.f,0.f,0.f,0.f,0.f,0.f};

    const int arow = tm * 16 + l16;
    const int bc0  = tn0 * 16 + l16;
    const int bc1  = tn1 * 16 + l16;

#pragma unroll 4
    for (int kb = 0; kb < 16; ++kb) {
        const int kbase = kb * 4 + half * 2;
        v2f a, b0, b1;
        {
            const float* p = A + arow * LSTR + kbase;
            a.x = p[0]; a.y = p[1];
        }
        if (TRANSB) {   // logical B = Bphys^T
            const float* p0 = B + bc0 * LSTR + kbase;
            const float* p1 = B + bc1 * LSTR + kbase;
            b0.x = p0[0]; b0.y = p0[1];
            b1.x = p1[0]; b1.y = p1[1];
        } else {
            const float* p = B + kbase * LSTR;
            b0.x = p[bc0]; b0.y = p[LSTR + bc0];
            b1.x = p[bc1]; b1.y = p[LSTR + bc1];
        }
        acc0 = __builtin_amdgcn_wmma_f32_16x16x4_f32(false, a, false, b0,
                                                     (short)0, acc0, false, false);
        acc1 = __builtin_amdgcn_wmma_f32_16x16x4_f32(false, a, false, b1,
                                                     (short)0, acc1, false, false);
    }

    __syncthreads();                         // all reads done -> in-place safe
#pragma unroll
    for (int r = 0; r < 8; ++r) {
        const int row = tm * 16 + r + 8 * half;
        float v0 = alpha * acc0[r];
        float v1 = alpha * acc1[r];
        if (row == bc0) v0 += beta_diag;
        if (row == bc1) v1 += beta_diag;
        D[row * LSTR + bc0] = v0;
        D[row * LSTR + bc1] = v1;
    }
    __syncthreads();
}

// Fused pair of independent 64x64x64 GEMMs in ONE barrier epoch:
//   waves 0-3: D0 = A0 @ B0      waves 4-7: D1 = A1 @ B1
// Each wave computes one 16-row tile stripe (4 tiles, shared A fragment).
// Halves the barrier count on the Newton-Schulz critical path.
__device__ __forceinline__ void gemm_pair(float* __restrict__ D0,
                                          const float* __restrict__ A0,
                                          const float* __restrict__ B0,
                                          float* __restrict__ D1,
                                          const float* __restrict__ A1,
                                          const float* __restrict__ B1)
{
    __syncthreads();
    const int lane = threadIdx.x & 31;
    const int wave = threadIdx.x >> 5;
    const int grp  = wave >> 2;              // uniform per wave -> EXEC stays full
    const int wq   = wave & 3;               // tile row for this wave
    const int half = lane >> 4;
    const int l16  = lane & 15;

    const float* A = grp ? A1 : A0;
    const float* B = grp ? B1 : B0;
    float*       D = grp ? D1 : D0;

    v8f acc[4];
#pragma unroll
    for (int t = 0; t < 4; ++t) acc[t] = (v8f){0.f,0.f,0.f,0.f,0.f,0.f,0.f,0.f};

    const int arow = wq * 16 + l16;

#pragma unroll 2
    for (int kb = 0; kb < 16; ++kb) {
        const int kbase = kb * 4 + half * 2;
        v2f a;
        {
            const float* p = A + arow * LSTR + kbase;
            a.x = p[0]; a.y = p[1];
        }
        const float* brow = B + kbase * LSTR + l16;
#pragma unroll
        for (int tn = 0; tn < 4; ++tn) {
            v2f b;
            b.x = brow[tn * 16];
            b.y = brow[tn * 16 + LSTR];
            acc[tn] = __builtin_amdgcn_wmma_f32_16x16x4_f32(false, a, false, b,
                                                            (short)0, acc[tn],
                                                            false, false);
        }
    }

    __syncthreads();                         // all reads (both products) done
#pragma unroll
    for (int tn = 0; tn < 4; ++tn) {
        const int col = tn * 16 + l16;
#pragma unroll
        for (int r = 0; r < 8; ++r) {
            const int row = wq * 16 + r + 8 * half;
            D[row * LSTR + col] = acc[tn][r];
        }
    }
    __syncthreads();
}

// ---- async global -> LDS staging (ASYNCcnt path, no VGPR round-trip) -------
__device__ __forceinline__ void load64_async_issue(float* dst,
                                                   const float* __restrict__ src)
{
    for (int i = threadIdx.x; i < NELEM / 4; i += 256) {
        const int r = i >> 4, c = (i & 15) * 4;
        const unsigned int lds = (unsigned int)(uintptr_t)(dst + r * LSTR + c);
        const float* g = src + i * 4;
        asm volatile("global_load_async_to_lds_b128 %0, %1, off"
                     :: "v"(lds), "v"(g) : "memory");
    }
}

__device__ __forceinline__ void load_async_finish()
{
    asm volatile("s_wait_asynccnt 0" ::: "memory");
    __syncthreads();
}

// ---- elementwise helpers (256 threads) -------------------------------------
__device__ __forceinline__ void set_identity(float* M)
{
    __syncthreads();
    for (int i = threadIdx.x; i < NELEM; i += 256) {
        const int r = i >> 6, c = i & 63;
        M[r * LSTR + c] = (r == c) ? 1.f : 0.f;
    }
    __syncthreads();
}

__device__ __forceinline__ void scale_inplace(float* M, float s)
{
    __syncthreads();
    for (int i = threadIdx.x; i < NELEM; i += 256) {
        const int r = i >> 6, c = i & 63;
        M[r * LSTR + c] *= s;
    }
    __syncthreads();
}

// Gershgorin / max-abs-row-sum bound: >= lambda_max for symmetric M.
__device__ __forceinline__ float spec_bound(const float* M, float* red)
{
    __syncthreads();
    if (threadIdx.x < NCDIM) {
        float s = 0.f;
        const float* row = M + threadIdx.x * LSTR;
        for (int j = 0; j < NCDIM; ++j) s += fabsf(row[j]);
        red[threadIdx.x] = s;
    }
    __syncthreads();
    float m = 1e-30f;
    for (int j = 0; j < NCDIM; ++j) m = fmaxf(m, red[j]);
    return m;
}

// ---------------------------------------------------------------------------
// Kernel 0 (one block): Cm12 = (S S^T + eps I)^(-1/2) via Newton-Schulz
// coupled inverse-sqrt iteration, written to global scratch.
// ---------------------------------------------------------------------------
__global__ __launch_bounds__(256)
void tangent_prep_cm12(const float* __restrict__ sqrt_cov, float* __restrict__ cm12)
{
    __shared__ float sC[NCDIM * LSTR];
    __shared__ float sX[NCDIM * LSTR];
    __shared__ float sZ[NCDIM * LSTR];
    __shared__ float sP[NCDIM * LSTR];
    __shared__ float red[NCDIM];

    load64_async_issue(sC, sqrt_cov);
    load_async_finish();

    gemm64<true>(sX, sC, sC, 1.f, EPSV);          // Cref = S S^T + eps I
    const float c = spec_bound(sX, red);
    scale_inplace(sX, 1.f / c);                   // spectrum in (0,1]
    set_identity(sZ);
    for (int it = 0; it < 12; ++it) {             // Y->Cn^(1/2), Z->Cn^(-1/2)
        gemm64<false>(sP, sZ, sX, -0.5f, 1.5f);   // P = 1.5 I - 0.5 Z Y
        gemm_pair(sX, sX, sP,                     // Y = Y P  (in place)
                  sZ, sP, sZ);                    // Z = P Z  (in place)
    }
    const float rs = rsqrtf(c);                   // (C/c)^(-1/2) * c^(-1/2) = C^(-1/2)
    for (int i = threadIdx.x; i < NELEM; i += 256) {
        const int r = i >> 6, cc = i & 63;
        cm12[i] = sZ[r * LSTR + cc] * rs;
    }
}

// ---------------------------------------------------------------------------
// Kernel 1 (one block per batch): whiten, matrix-log via inverse scaling &
// squaring (4 Newton-Schulz sqrts + 4-term Mercator series), scale + triu.
// ---------------------------------------------------------------------------
__global__ __launch_bounds__(256)
void tangent_batch(const float* __restrict__ x, const float* __restrict__ cm12,
                   const float* __restrict__ noise, float* __restrict__ out)
{
    __shared__ float sC[NCDIM * LSTR];
    __shared__ float sX[NCDIM * LSTR];
    __shared__ float sY[NCDIM * LSTR];
    __shared__ float sZ[NCDIM * LSTR];
    __shared__ float sP[NCDIM * LSTR];
    __shared__ float red[NCDIM];

    const int b = blockIdx.x;
    load64_async_issue(sC, cm12);                 // both transfers in flight,
    load64_async_issue(sX, x + (size_t)b * NELEM);// one wait
    load_async_finish();

    gemm64<false>(sY, sC, sX, 1.f, 0.f);          // T1 = Cm12 x
    gemm64<false>(sX, sY, sC, 1.f, 0.f);          // m  = T1 Cm12

    if (threadIdx.x < NCDIM)                      // m += eps*diag(noise)
        sX[threadIdx.x * LSTR + threadIdx.x] += EPSV * noise[threadIdx.x];

    const float c  = spec_bound(sX, red);         // also syncs the diag update
    const float lc = logf(c);
    scale_inplace(sX, 1.f / c);                   // A0 = m/c, spectrum (0,1]

    // A0^(1/16): 4 chained coupled Newton-Schulz square roots, in place in sX
    for (int s = 0; s < SQRT_STAGES; ++s) {
        set_identity(sZ);
        for (int it = 0; it < NS_ITERS; ++it) {
            gemm64<false>(sP, sZ, sX, -0.5f, 1.5f);
            gemm_pair(sX, sX, sP,                 // Y = Y P  (in place)
                      sZ, sP, sZ);                // Z = P Z  (in place)
        }
    }

    // E = Y16 - I ; log(A0) ~= 16 * (E - E^2/2 + E^3/3 - E^4/4)
    __syncthreads();
    if (threadIdx.x < NCDIM)
        sX[threadIdx.x * LSTR + threadIdx.x] -= 1.f;
    gemm64<false>(sP, sX, sX, 1.f, 0.f);          // E^2
    gemm_pair(sY, sP, sX,                         // E^3
              sZ, sP, sP);                        // E^4  (independent -> fused)

    // triu extraction with sqrt(2) off-diag coeffs; log m = lc*I + 16*series
    const float SQ2 = 1.41421356237309515f;
    for (int e = threadIdx.x; e < NTRI; e += 256) {
        int i = 0, base = 0;
        while (base + (NCDIM - i) <= e) { base += NCDIM - i; ++i; }
        const int j   = i + (e - base);
        const int off = i * LSTR + j;
        float v = 16.f * (sX[off] - 0.5f * sP[off]
                          + (1.f / 3.f) * sY[off] - 0.25f * sZ[off]);
        if (i == j) v += lc;
        out[(size_t)b * NTRI + e] = (i == j) ? v : SQ2 * v;
    }
}

// ---------------------------------------------------------------------------
extern "C" void kernel_launch(void* const* d_in, const int* in_sizes, int n_in,
                              void* d_out, int out_size, void* d_ws, size_t ws_size,
                              hipStream_t stream)
{
    const float* x        = (const float*)d_in[0];   // (NT, 64, 64) f32
    const float* sqrt_cov = (const float*)d_in[1];   // (64, 64)     f32
    const float* noise    = (const float*)d_in[2];   // (64,)        f32
    float*       out      = (float*)d_out;           // (NT, 2080)   f32
    float*       cm12     = (float*)d_ws;            // 4096 floats of scratch

    const int nt = in_sizes[0] / NELEM;

    tangent_prep_cm12<<<1, 256, 0, stream>>>(sqrt_cov, cm12);
    tangent_batch<<<nt, 256, 0, stream>>>(x, cm12, noise, out);
}